// _NNConv_16784732193361
// MI455X (gfx1250) — compile-verified
//
#include <hip/hip_runtime.h>

typedef __attribute__((ext_vector_type(2))) float v2f;
typedef __attribute__((ext_vector_type(4))) float v4f;
typedef __attribute__((ext_vector_type(8))) float v8f;

#define IN_DIM  16
#define OUT_DIM 16
#define EDGE_IN 16
#define HID     32

// LDS row strides (dwords), chosen so 16-lane half-wave accesses are bank-conflict-free
// (stride mod 64: 20 -> 16 distinct banks; 34 -> 32 distinct; 532 mod 64 = 20 -> distinct)
#define W1T_STRIDE 20    // w1T[32][20]  : w1T[n][k] = w1[k*32+n]
#define W2T_STRIDE 532   // w2T[16][532] : cols 0..511 = w2 (c=16k+i), cols 512..527 = b2
#define HB_STRIDE  34    // hbuf[16][34] : h[e][n]
#define XB_STRIDE  20    // xbuf[16][20] : x_src[e][i]

static __device__ __forceinline__ v8f wmma4(v2f a, v2f b, v8f c) {
  // D = A(16x4 f32) * B(4x16 f32) + C(16x16 f32)
  return __builtin_amdgcn_wmma_f32_16x16x4_f32(false, a, false, b, (short)0, c,
                                               false, false);
}

static __device__ __forceinline__ void atomAddF(float* p, float v) {
  unsafeAtomicAdd(p, v);   // global_atomic_add_f32
}

__global__ void __launch_bounds__(256)
nnconv_edge_kernel(const float* __restrict__ x, const int* __restrict__ src,
                   const int* __restrict__ dst, const float* __restrict__ ea,
                   const float* __restrict__ w1, const float* __restrict__ b1,
                   const float* __restrict__ w2, const float* __restrict__ b2,
                   float* __restrict__ out_sum, float* __restrict__ cnt,
                   int n_tiles) {
  __shared__ float w1T[32 * W1T_STRIDE];
  __shared__ float w2T[16 * W2T_STRIDE];
  __shared__ float hbuf[8][16 * HB_STRIDE];
  __shared__ float xbuf[8][16 * XB_STRIDE];
  __shared__ int   dstbuf[8][16];

  const int tid = threadIdx.x;

  // ---- Phase 0: cooperative fill of shared B matrices (transposed) ----
  for (int j = tid; j < 16 * 32; j += 256) {          // w1 is [16][32] row-major
    int k = j >> 5, n = j & 31;
    w1T[n * W1T_STRIDE + k] = w1[j];
  }
  for (int j = tid; j < 512 * 16; j += 256) {         // w2 flat [32*256]; c=j>>4, o=j&15
    w2T[(j & 15) * W2T_STRIDE + (j >> 4)] = w2[j];
  }
  {                                                    // b2 flat [256]; i=j>>4, o=j&15
    int j = tid;
    if (j < 256) w2T[(j & 15) * W2T_STRIDE + 512 + (j >> 4)] = b2[j];
  }
  __syncthreads();

  const int lane = tid & 31;
  const int wave = tid >> 5;
  const int half = lane >> 4;     // which 16-lane half
  const int row  = lane & 15;     // M-row (edge in tile) == N-col (output feature)
  const int tile = blockIdx.x * 8 + wave;
  if (tile >= n_tiles) return;
  const long ebase = (long)tile * 16;

  // ---- Phase 1a: gather src rows of x + dst indices into LDS ----
  if (lane < 16) {
    const int e = lane;
    const int si = src[ebase + e];
    dstbuf[wave][e] = dst[ebase + e];
    const float* xr = x + (long)si * IN_DIM;
#pragma unroll
    for (int q = 0; q < 4; ++q)
      *(v4f*)&xbuf[wave][e * XB_STRIDE + 4 * q] = *(const v4f*)(xr + 4 * q);
  }

  // A fragments of edge_attr tile: lane holds row (L%16), K pair (4s + 2*half)
  v2f eaf[4];
#pragma unroll
  for (int s = 0; s < 4; ++s)
    eaf[s] = *(const v2f*)(ea + (ebase + row) * EDGE_IN + 4 * s + 2 * half);

  // ---- Phase 1b: h = relu(ea @ w1 + b1)  -> two 16x16 C tiles, K=16 ----
  v8f c0 = {}, c1 = {};
#pragma unroll
  for (int s = 0; s < 4; ++s) {
    v2f bA = *(const v2f*)&w1T[row * W1T_STRIDE + 4 * s + 2 * half];
    v2f bB = *(const v2f*)&w1T[(16 + row) * W1T_STRIDE + 4 * s + 2 * half];
    c0 = wmma4(eaf[s], bA, c0);
    c1 = wmma4(eaf[s], bB, c1);
  }
  const float bb0 = b1[row], bb1 = b1[16 + row];
#pragma unroll
  for (int v = 0; v < 8; ++v) {
    int e = v + 8 * half;                                   // C layout: M = v + 8*half
    hbuf[wave][e * HB_STRIDE + row]      = fmaxf(c0[v] + bb0, 0.f);
    hbuf[wave][e * HB_STRIDE + 16 + row] = fmaxf(c1[v] + bb1, 0.f);
  }

  // ---- Phase 2: msg = P[16,528] @ W2cat[528,16], A built on the fly ----
  // p[e][c] = h[e][c>>4] * x[e][c&15] for c<512 ; = x[e][c-512] for c>=512 (bias).
  // Key invariances: within a k-group of 4 steps, h[e][k] is constant, and the
  // x pair index i = 4q + 2*half depends only on q -> hoist all 4 x pairs into
  // registers for the entire K=528 loop. Two accumulators break the serial
  // WMMA->WMMA accumulation chain.
  v2f xp[4];
#pragma unroll
  for (int q = 0; q < 4; ++q)
    xp[q] = *(const v2f*)&xbuf[wave][row * XB_STRIDE + 4 * q + 2 * half];

  const float* hrow  = &hbuf[wave][row * HB_STRIDE];
  const float* w2row = &w2T[row * W2T_STRIDE + 2 * half];

  v8f acc0 = {}, acc1 = {};
#pragma unroll 8
  for (int k = 0; k < HID; ++k) {
    const float hv = hrow[k];
    const float* wk = w2row + 16 * k;
#pragma unroll
    for (int q = 0; q < 4; ++q) {
      v2f a; a.x = hv * xp[q].x; a.y = hv * xp[q].y;
      const v2f b = *(const v2f*)(wk + 4 * q);
      if (q & 1) acc1 = wmma4(a, b, acc1);
      else       acc0 = wmma4(a, b, acc0);
    }
  }
  // Folded bias rows: p[e][512+i] = x[e][i] -> reuse hoisted xp[q] directly.
#pragma unroll
  for (int q = 0; q < 4; ++q) {
    const v2f b = *(const v2f*)(w2row + 512 + 4 * q);
    if (q & 1) acc1 = wmma4(xp[q], b, acc1);
    else       acc0 = wmma4(xp[q], b, acc0);
  }

  // ---- Phase 3: scatter-add (L2-resident atomics) ----
#pragma unroll
  for (int v = 0; v < 8; ++v) {
    const int e = v + 8 * half;
    const int d = dstbuf[wave][e];
    atomAddF(&out_sum[(long)d * OUT_DIM + row], acc0[v] + acc1[v]);
  }
  if (lane < 16) atomAddF(&cnt[dstbuf[wave][lane]], 1.0f);
}

__global__ void zero_kernel(float* __restrict__ out, float* __restrict__ cnt,
                            int nout, int ncnt) {
  int i = blockIdx.x * blockDim.x + threadIdx.x;
  if (i < nout) out[i] = 0.f;
  if (i < ncnt) cnt[i] = 0.f;
}

// Scalar fallback for a partial tail tile (E % 16 != 0); not launched for E=800000.
__global__ void nnconv_tail_kernel(const float* __restrict__ x, const int* __restrict__ src,
                                   const int* __restrict__ dst, const float* __restrict__ ea,
                                   const float* __restrict__ w1, const float* __restrict__ b1,
                                   const float* __restrict__ w2, const float* __restrict__ b2,
                                   float* __restrict__ out_sum, float* __restrict__ cnt,
                                   int estart, int E) {
  int e = estart + blockIdx.x * blockDim.x + threadIdx.x;
  if (e >= E) return;
  float h[HID];
  for (int n = 0; n < HID; ++n) {
    float s = b1[n];
    for (int k = 0; k < EDGE_IN; ++k) s += ea[(long)e * EDGE_IN + k] * w1[k * HID + n];
    h[n] = fmaxf(s, 0.f);
  }
  const int si = src[e], di = dst[e];
  for (int o = 0; o < OUT_DIM; ++o) {
    float m = 0.f;
    for (int i = 0; i < IN_DIM; ++i) {
      float wio = b2[i * OUT_DIM + o];
      for (int k = 0; k < HID; ++k) wio += h[k] * w2[k * IN_DIM * OUT_DIM + i * OUT_DIM + o];
      m += x[(long)si * IN_DIM + i] * wio;
    }
    atomAddF(&out_sum[(long)di * OUT_DIM + o], m);
  }
  atomAddF(&cnt[di], 1.0f);
}

__global__ void finalize_kernel(float* __restrict__ out, const float* __restrict__ cnt,
                                const float* __restrict__ x, const float* __restrict__ root,
                                const float* __restrict__ bias, int N) {
  int idx = blockIdx.x * blockDim.x + threadIdx.x;
  if (idx >= N * OUT_DIM) return;
  const int n = idx >> 4, o = idx & 15;
  float c = cnt[n];
  c = c > 1.f ? c : 1.f;
  float r = out[idx] / c + bias[o];
#pragma unroll
  for (int i = 0; i < IN_DIM; ++i) r += x[n * IN_DIM + i] * root[i * OUT_DIM + o];
  out[idx] = r;
}

extern "C" void kernel_launch(void* const* d_in, const int* in_sizes, int n_in,
                              void* d_out, int out_size, void* d_ws, size_t ws_size,
                              hipStream_t stream) {
  const float* x    = (const float*)d_in[0];
  const int*   eidx = (const int*)d_in[1];
  const float* ea   = (const float*)d_in[2];
  const float* w1   = (const float*)d_in[3];
  const float* b1   = (const float*)d_in[4];
  const float* w2   = (const float*)d_in[5];
  const float* b2   = (const float*)d_in[6];
  const float* root = (const float*)d_in[7];
  const float* bias = (const float*)d_in[8];

  const int N = in_sizes[0] / IN_DIM;
  const int E = in_sizes[1] / 2;
  const int* src = eidx;
  const int* dst = eidx + E;

  float* out = (float*)d_out;
  float* cnt = (float*)d_ws;     // N floats of scratch for degree counts

  zero_kernel<<<(N * OUT_DIM + 255) / 256, 256, 0, stream>>>(out, cnt, N * OUT_DIM, N);

  const int n_tiles = E / 16;
  if (n_tiles > 0) {
    const int blocks = (n_tiles + 7) / 8;   // 8 waves (tiles) per 256-thread block
    nnconv_edge_kernel<<<blocks, 256, 0, stream>>>(x, src, dst, ea, w1, b1, w2, b2,
                                                   out, cnt, n_tiles);
  }
  const int tail_start = n_tiles * 16;
  if (E - tail_start > 0)
    nnconv_tail_kernel<<<(E - tail_start + 255) / 256, 256, 0, stream>>>(
        x, src, dst, ea, w1, b1, w2, b2, out, cnt, tail_start, E);

  finalize_kernel<<<(N * OUT_DIM + 255) / 256, 256, 0, stream>>>(out, cnt, x, root, bias, N);
}